// SymbioseGNN_27685359190830
// MI455X (gfx1250) — compile-verified
//
#include <hip/hip_runtime.h>
#include <hip/hip_bf16.h>

typedef float v2f __attribute__((ext_vector_type(2)));
typedef float v4f __attribute__((ext_vector_type(4)));
typedef float v8f __attribute__((ext_vector_type(8)));

#define IN_DIM  256
#define HID_DIM 128
#define OUT_DIM 64

// ---------------- small elementwise kernels ----------------

__global__ void k_set1(float* p, int n) {
    int i = blockIdx.x * blockDim.x + threadIdx.x;
    if (i < n) p[i] = 1.0f;
}

__global__ void k_degcount(const long long* __restrict__ dst, float* deg, int e) {
    int i = blockIdx.x * blockDim.x + threadIdx.x;
    if (i < e) atomicAdd(&deg[(int)dst[i]], 1.0f);
}

__global__ void k_rsqrt_inplace(float* p, int n) {
    int i = blockIdx.x * blockDim.x + threadIdx.x;
    if (i < n) p[i] = rsqrtf(p[i]);
}

// agg[i, :] = hw[i, :] * dinv[i]^2 + bias[:]   (self-loop message + bias), float4-wide
__global__ void k_selfbias4(const v4f* __restrict__ hw, const float* __restrict__ dinv,
                            const v4f* __restrict__ bias, v4f* __restrict__ agg,
                            long total4, int D4) {
    long i = (long)blockIdx.x * blockDim.x + threadIdx.x;
    if (i >= total4) return;
    int node = (int)(i / D4);
    int j    = (int)(i % D4);
    float dv = dinv[node];
    float s  = dv * dv;
    v4f h = hw[i];
    v4f b = bias[j];
    v4f r;
    r.x = h.x * s + b.x;
    r.y = h.y * s + b.y;
    r.z = h.z * s + b.z;
    r.w = h.w * s + b.w;
    agg[i] = r;
}

__global__ void k_relu4_inplace(v4f* p, long n4) {
    long i = (long)blockIdx.x * blockDim.x + threadIdx.x;
    if (i >= n4) return;
    v4f v = p[i];
    v.x = fmaxf(v.x, 0.0f);
    v.y = fmaxf(v.y, 0.0f);
    v.z = fmaxf(v.z, 0.0f);
    v.w = fmaxf(v.w, 0.0f);
    p[i] = v;
}

// ---------------- edge scatter: one wave (32 lanes) per edge ----------------
// Each lane owns D/32 consecutive feature channels -> one vector load + VW atomics.

template<int D>
__global__ void k_edge_agg(const long long* __restrict__ src, const long long* __restrict__ dst,
                           const float* __restrict__ dinv, const float* __restrict__ hw,
                           float* __restrict__ agg, int nedges) {
    constexpr int VW = D / 32;                       // 4 (D=128) or 2 (D=64)
    typedef float vec_t __attribute__((ext_vector_type(VW)));
    const int wavesPerBlock = blockDim.x >> 5;
    const int w    = blockIdx.x * wavesPerBlock + (threadIdx.x >> 5);
    const int lane = threadIdx.x & 31;
    if (w >= nedges) return;
    const int s = (int)src[w];
    const int d = (int)dst[w];
    const float norm = dinv[s] * dinv[d];
    const vec_t v = *(const vec_t*)(hw + (size_t)s * D + lane * VW);
    float* ad = agg + (size_t)d * D + lane * VW;
#pragma unroll
    for (int k = 0; k < VW; ++k) {
        atomicAdd(&ad[k], v[k] * norm);
    }
}

// ---------------- fp32 WMMA GEMM: Y[N,DOUT] = X[N,K] @ W[K,DOUT] ----------------
// One 16x16 output tile per wave; DOUT/16 waves per block cover one 16-row stripe.
// V_WMMA_F32_16X16X4_F32; A-tile staged in LDS.
// Lane layouts per CDNA5 ISA 7.12.2:
//   A 16x4 f32 : lane t -> row (t&15), VGPR pair holds K = (t>>4)*2 .. +1
//   B 4x16 f32 : lane t -> col (t&15), VGPR pair holds K = (t>>4)*2 .. +1
//   C/D 16x16  : VGPR r -> row r + 8*(t>>4), col (t&15)

template<int K, int DOUT>
__global__ __launch_bounds__(32 * (DOUT / 16))
void k_gemm_wmma_f32(const float* __restrict__ X, const float* __restrict__ W,
                     float* __restrict__ Y, int nrows) {
    constexpr int NT = DOUT / 16;
    __shared__ float As[16 * K];

    const int m0 = blockIdx.x * 16;
    const bool full = (m0 + 16 <= nrows);            // uniform across block

    if (full) {
        // unguarded float4 staging; a float4 never crosses a row (K % 4 == 0)
        for (int i = 4 * threadIdx.x; i < 16 * K; i += 4 * 32 * NT) {
            const int r = i / K;
            const int c = i % K;
            *(v4f*)&As[i] = *(const v4f*)&X[(size_t)(m0 + r) * K + c];
        }
    } else {
        for (int i = threadIdx.x; i < 16 * K; i += 32 * NT) {
            const int r = i / K;
            const int c = i % K;
            const int row = m0 + r;
            As[i] = (row < nrows) ? X[(size_t)row * K + c] : 0.0f;
        }
    }
    __syncthreads();

    const int lane = threadIdx.x & 31;
    const int wv   = threadIdx.x >> 5;
    const int mr   = lane & 15;          // A row / B col within tile
    const int hi   = lane >> 4;          // lane-half select
    const int kh   = hi * 2;             // K sub-pair (0 or 2)
    const int n    = wv * 16 + mr;       // global output column

    const float* wp = W + (size_t)kh * DOUT + n;     // B column base for this lane

    v8f c = {};
#pragma unroll 8
    for (int k0 = 0; k0 < K; k0 += 4) {
        v2f a = *(const v2f*)&As[mr * K + k0 + kh];
        v2f b;
        b.x = wp[(size_t)k0 * DOUT];
        b.y = wp[(size_t)(k0 + 1) * DOUT];
        c = __builtin_amdgcn_wmma_f32_16x16x4_f32(
                /*neg_a=*/false, a, /*neg_b=*/false, b,
                /*c_mod=*/(short)0, c, /*reuse_a=*/false, /*reuse_b=*/false);
    }

    float* yp = Y + (size_t)(m0 + hi * 8) * DOUT + n;
    if (full) {
#pragma unroll
        for (int r = 0; r < 8; ++r) yp[r * DOUT] = c[r];
    } else {
#pragma unroll
        for (int r = 0; r < 8; ++r) {
            if (m0 + hi * 8 + r < nrows) yp[r * DOUT] = c[r];
        }
    }
}

// ---------------- driver ----------------

extern "C" void kernel_launch(void* const* d_in, const int* in_sizes, int n_in,
                              void* d_out, int out_size, void* d_ws, size_t ws_size,
                              hipStream_t stream) {
    const float*     x    = (const float*)d_in[0];
    const long long* ei   = (const long long*)d_in[1];   // int64 edge_index, [2, E] flat
    const float*     W1   = (const float*)d_in[2];
    const float*     b1   = (const float*)d_in[3];
    const float*     W2   = (const float*)d_in[4];
    const float*     b2   = (const float*)d_in[5];

    const int N = in_sizes[0] / IN_DIM;
    const int E = in_sizes[1] / 2;
    const long long* srcp = ei;       // edge_index[0]
    const long long* dstp = ei + E;   // edge_index[1]

    char* ws = (char*)d_ws;
    float* hw   = (float*)ws;                                   // N*HID (reused N*OUT for layer 2)
    float* aggh = (float*)(ws + (size_t)N * HID_DIM * 4);       // N*HID  (agg1 -> h)
    float* dinv = (float*)(ws + (size_t)N * HID_DIM * 8);       // N      (deg -> dinv)
    float* out  = (float*)d_out;                                // N*OUT

    const int TPB = 256;
    const int mtiles = (N + 15) / 16;

    // --- degrees & symmetric norm ---
    k_set1<<<(N + TPB - 1) / TPB, TPB, 0, stream>>>(dinv, N);
    k_degcount<<<(E + TPB - 1) / TPB, TPB, 0, stream>>>(dstp, dinv, E);
    k_rsqrt_inplace<<<(N + TPB - 1) / TPB, TPB, 0, stream>>>(dinv, N);

    // --- layer 1: hw1 = x @ W1 ---
    k_gemm_wmma_f32<IN_DIM, HID_DIM><<<mtiles, 32 * (HID_DIM / 16), 0, stream>>>(x, W1, hw, N);

    // agg1 = hw1*dinv^2 + b1 ; += edge messages ; relu
    {
        long total4 = (long)N * HID_DIM / 4;
        k_selfbias4<<<(int)((total4 + TPB - 1) / TPB), TPB, 0, stream>>>(
            (const v4f*)hw, dinv, (const v4f*)b1, (v4f*)aggh, total4, HID_DIM / 4);
        const int wpb = TPB / 32;
        k_edge_agg<HID_DIM><<<(E + wpb - 1) / wpb, TPB, 0, stream>>>(srcp, dstp, dinv, hw, aggh, E);
        k_relu4_inplace<<<(int)((total4 + TPB - 1) / TPB), TPB, 0, stream>>>((v4f*)aggh, total4);
    }

    // --- layer 2: hw2 = h @ W2  (reuse hw buffer) ---
    k_gemm_wmma_f32<HID_DIM, OUT_DIM><<<mtiles, 32 * (OUT_DIM / 16), 0, stream>>>(aggh, W2, hw, N);

    // out = hw2*dinv^2 + b2 ; += edge messages
    {
        long total4 = (long)N * OUT_DIM / 4;
        k_selfbias4<<<(int)((total4 + TPB - 1) / TPB), TPB, 0, stream>>>(
            (const v4f*)hw, dinv, (const v4f*)b2, (v4f*)out, total4, OUT_DIM / 4);
        const int wpb = TPB / 32;
        k_edge_agg<OUT_DIM><<<(E + wpb - 1) / wpb, TPB, 0, stream>>>(srcp, dstp, dinv, hw, out, E);
    }
}